// SupportLowRankRNN_91036126806945
// MI455X (gfx1250) — compile-verified
//
#include <hip/hip_runtime.h>
#include <stdint.h>

#define HIDDEN     1024
#define RANK       2
#define IN_SIZE    4
#define OUT_SIZE   2
#define N_SUPPORTS 2
#define GBD        8
#define SEQ        2048
#define BATCH      32
#define NOISE_STD  0.05f
#define ALPHA      0.2f

#define THREADS 256
#define EPT     (HIDDEN / THREADS)   // 4 elements per thread
#define NWAVES  (THREADS / 32)       // 8 wave32s
#define RING    4                    // noise prefetch depth (power of 2)
#define PROW    12                   // padded param row (floats) -> 48B, b128-aligned

static_assert(EPT * THREADS == HIDDEN, "layout");

// ---- CDNA5 async global->LDS path (probe-confirmed this toolchain) ----
#if defined(__has_builtin)
#  if __has_builtin(__builtin_amdgcn_global_load_async_to_lds_b128)
#    define HAVE_ASYNC 1
#  endif
#  if __has_builtin(__builtin_amdgcn_wmma_f32_16x16x4_f32)
#    define HAVE_WMMA_F32 1
#  endif
#endif
#ifndef HAVE_ASYNC
#  define HAVE_ASYNC 0
#endif
#ifndef HAVE_WMMA_F32
#  define HAVE_WMMA_F32 0
#endif

// Exact pointee type per hipcc diagnostic: 'int __vector_size__(4*sizeof(int))'
typedef int v4i_ __attribute__((vector_size(16)));
typedef __attribute__((address_space(1))) v4i_ g_v4i;   // global
typedef __attribute__((address_space(3))) v4i_ l_v4i;   // LDS

typedef float v2f __attribute__((ext_vector_type(2)));
typedef float v8f __attribute__((ext_vector_type(8)));

__device__ __forceinline__ void async_copy_f4(const float* g, float* l) {
#if HAVE_ASYNC
  __builtin_amdgcn_global_load_async_to_lds_b128(
      (g_v4i*)(const_cast<float*>(g)), (l_v4i*)l, 0, 0);
#else
  *(float4*)l = *(const float4*)g;   // synchronous fallback
#endif
}

#if HAVE_ASYNC
#  if defined(__has_builtin) && __has_builtin(__builtin_amdgcn_s_wait_asynccnt)
#    define WAIT_ASYNC_LE(n) __builtin_amdgcn_s_wait_asynccnt(n)
#  else
#    define WAIT_ASYNC_LE(n) asm volatile("s_wait_asynccnt %0" :: "n"(n) : "memory")
#  endif
#else
#  define WAIT_ASYNC_LE(n) do {} while (0)
#endif

// Packed param matrix row m, support s, basis dim d:
// rows 0-3: wi[k=m], 4-5: m_rec[m-4], 6-7: n_rec[m-6], 8-9: wo[m-8], 10-15: 0
__device__ __forceinline__ float pval(int m, int s, int d,
                                      const float* wi, const float* mp,
                                      const float* np, const float* wo) {
  float v = 0.f;
  if (m < 4)       v = wi[((m    ) * N_SUPPORTS + s) * GBD + d];
  else if (m < 6)  v = mp[((m - 4) * N_SUPPORTS + s) * GBD + d];
  else if (m < 8)  v = np[((m - 6) * N_SUPPORTS + s) * GBD + d];
  else if (m < 10) v = wo[((m - 8) * N_SUPPORTS + s) * GBD + d];
  return v;
}

__global__ __launch_bounds__(THREADS, 1)
void SupportLowRankRNN_kernel(const float* __restrict__ input,     // [B,SEQ,IN]
                              const float* __restrict__ noise,     // [B,SEQ,H]
                              const float* __restrict__ basis,     // [GBD,H]
                              const float* __restrict__ supports,  // [NS,H]
                              const float* __restrict__ wi,        // [IN,NS,GBD]
                              const float* __restrict__ mp,        // [RANK,NS,GBD]
                              const float* __restrict__ np,        // [RANK,NS,GBD]
                              const float* __restrict__ wo,        // [OUT,NS,GBD]
                              const float* __restrict__ h0,        // [H]
                              float* __restrict__ out)             // [B,SEQ,OUT]
{
  // Staging scratch for the WMMA precompute exactly overlays the streaming
  // buffers (both 48 KB); staging is finished (barrier) before DMA starts.
  __shared__ union {
    struct {
      float  input[SEQ * IN_SIZE];        // 32 KB: whole input row
      float4 noise[RING][THREADS];        // 16 KB: noise ring
    } run;
    float params[HIDDEN][PROW];           // 48 KB: 10 params/column, padded
  } sm;
  __shared__ float lds_red[2][NWAVES][4]; // double-buffered partials

  const int b    = blockIdx.x;
  const int tid  = threadIdx.x;
  const int lane = tid & 31;
  const int wave = tid >> 5;
  const int j0   = tid * EPT;

  const size_t noise_base = (size_t)b * SEQ * HIDDEN + (size_t)j0;
  const float* inb  = input + (size_t)b * SEQ * IN_SIZE;
  float*       outp = out   + (size_t)b * SEQ * OUT_SIZE;

  // ---- 1) proxy-parameter build ----
  float wiT[EPT][IN_SIZE], mT[EPT][RANK], nT[EPT][RANK], woT[EPT][OUT_SIZE];
  const float inv_h = 1.0f / (float)HIDDEN;

#if HAVE_WMMA_F32
  {
    // D = P(16x8) x basis(8x16-tile) via two K=4 f32 WMMAs, per support.
    const int lhalf = lane >> 4;     // 0: lanes 0-15, 1: lanes 16-31
    const int mrow  = lane & 15;

    // A-matrix (16x4 f32): VGPR0 holds K=0|K=2 across lane halves, VGPR1 K=1|K=3.
    v2f Ab[N_SUPPORTS][2];
    #pragma unroll
    for (int s = 0; s < N_SUPPORTS; ++s) {
      #pragma unroll
      for (int kh = 0; kh < 2; ++kh) {
        const int kb = kh * 4 + lhalf * 2;
        Ab[s][kh][0] = pval(mrow, s, kb + 0, wi, mp, np, wo);
        Ab[s][kh][1] = pval(mrow, s, kb + 1, wi, mp, np, wo);
      }
    }

    #pragma unroll
    for (int c = 0; c < HIDDEN / (NWAVES * 16); ++c) {   // 8 16-col tiles per wave
      const int jb  = wave * (HIDDEN / NWAVES) + c * 16;
      const int col = jb + mrow;
      // B-matrix (4x16 f32): VGPR0 K=0|K=2, VGPR1 K=1|K=3; N = lane&15.
      v2f Blo, Bhi;
      Blo[0] = basis[(0 + lhalf * 2) * HIDDEN + col];
      Blo[1] = basis[(1 + lhalf * 2) * HIDDEN + col];
      Bhi[0] = basis[(4 + lhalf * 2) * HIDDEN + col];
      Bhi[1] = basis[(5 + lhalf * 2) * HIDDEN + col];

      v8f a0 = {}, a1 = {};
      a0 = __builtin_amdgcn_wmma_f32_16x16x4_f32(false, Ab[0][0], false, Blo,
                                                 (short)0, a0, false, false);
      a0 = __builtin_amdgcn_wmma_f32_16x16x4_f32(false, Ab[0][1], false, Bhi,
                                                 (short)0, a0, false, false);
      a1 = __builtin_amdgcn_wmma_f32_16x16x4_f32(false, Ab[1][0], false, Blo,
                                                 (short)0, a1, false, false);
      a1 = __builtin_amdgcn_wmma_f32_16x16x4_f32(false, Ab[1][1], false, Bhi,
                                                 (short)0, a1, false, false);

      // Support-mask combine (per column), then stage D to LDS.
      const float s0 = supports[col], s1 = supports[HIDDEN + col];
      float comb[8];
      #pragma unroll
      for (int i = 0; i < 8; ++i) comb[i] = s0 * a0[i] + s1 * a1[i];

      if (lane < 16) {           // rows 0..7 of this column
        float4 lo; lo.x = comb[0]; lo.y = comb[1]; lo.z = comb[2]; lo.w = comb[3];
        float4 hi; hi.x = comb[4]; hi.y = comb[5]; hi.z = comb[6]; hi.w = comb[7];
        *(float4*)&sm.params[col][0] = lo;
        *(float4*)&sm.params[col][4] = hi;
      } else {                   // rows 8..9 (rows 10..15 are zero padding)
        sm.params[col][8] = comb[0];
        sm.params[col][9] = comb[1];
      }
    }
  }
  __syncthreads();               // publish staged params
  #pragma unroll
  for (int e = 0; e < EPT; ++e) {
    const int j = j0 + e;
    #pragma unroll
    for (int k = 0; k < IN_SIZE; ++k)  wiT[e][k] = ALPHA * sm.params[j][k];
    #pragma unroll
    for (int r = 0; r < RANK; ++r) {
      mT[e][r] = sm.params[j][4 + r];
      nT[e][r] = (ALPHA * inv_h) * sm.params[j][6 + r];
    }
    #pragma unroll
    for (int o = 0; o < OUT_SIZE; ++o) woT[e][o] = inv_h * sm.params[j][8 + o];
  }
  __syncthreads();               // all reads done before DMA reuses this LDS
#else
  // Scalar fallback precompute
  #pragma unroll
  for (int e = 0; e < EPT; ++e) {
    const int j = j0 + e;
    float g[GBD];
    #pragma unroll
    for (int d = 0; d < GBD; ++d) g[d] = basis[d * HIDDEN + j];
    const float s0 = supports[j];
    const float s1 = supports[HIDDEN + j];
    #pragma unroll
    for (int k = 0; k < IN_SIZE; ++k) {
      float a0 = 0.f, a1 = 0.f;
      #pragma unroll
      for (int d = 0; d < GBD; ++d) {
        a0 += wi[(k * N_SUPPORTS + 0) * GBD + d] * g[d];
        a1 += wi[(k * N_SUPPORTS + 1) * GBD + d] * g[d];
      }
      wiT[e][k] = ALPHA * (s0 * a0 + s1 * a1);
    }
    #pragma unroll
    for (int r = 0; r < RANK; ++r) {
      float a0 = 0.f, a1 = 0.f, c0 = 0.f, c1 = 0.f;
      #pragma unroll
      for (int d = 0; d < GBD; ++d) {
        a0 += mp[(r * N_SUPPORTS + 0) * GBD + d] * g[d];
        a1 += mp[(r * N_SUPPORTS + 1) * GBD + d] * g[d];
        c0 += np[(r * N_SUPPORTS + 0) * GBD + d] * g[d];
        c1 += np[(r * N_SUPPORTS + 1) * GBD + d] * g[d];
      }
      mT[e][r] = s0 * a0 + s1 * a1;
      nT[e][r] = (ALPHA * inv_h) * (s0 * c0 + s1 * c1);
    }
    #pragma unroll
    for (int o = 0; o < OUT_SIZE; ++o) {
      float a0 = 0.f, a1 = 0.f;
      #pragma unroll
      for (int d = 0; d < GBD; ++d) {
        a0 += wo[(o * N_SUPPORTS + 0) * GBD + d] * g[d];
        a1 += wo[(o * N_SUPPORTS + 1) * GBD + d] * g[d];
      }
      woT[e][o] = inv_h * (s0 * a0 + s1 * a1);
    }
  }
#endif

  // ---- 2) kick off async DMA: whole input row + first RING noise slabs ----
  #pragma unroll
  for (int it = 0; it < (SEQ * IN_SIZE / 4) / THREADS; ++it) {   // exactly 8
    const int i = it * THREADS + tid;
    async_copy_f4(inb + i * 4, &sm.run.input[i * 4]);
  }
  #pragma unroll
  for (int tt = 0; tt < RING; ++tt)
    async_copy_f4(noise + noise_base + (size_t)tt * HIDDEN,
                  (float*)&sm.run.noise[tt][tid]);

  // ---- 3) state init + prologue reduction (th = tanh(h0)) ----
  float h[EPT], th[EPT];
  {
    float4 h4 = ((const float4*)h0)[tid];
    h[0] = h4.x; h[1] = h4.y; h[2] = h4.z; h[3] = h4.w;
  }
  #pragma unroll
  for (int e = 0; e < EPT; ++e) th[e] = tanhf(h[e]);

  {
    float p0 = 0.f, p1 = 0.f, p2 = 0.f, p3 = 0.f;
    #pragma unroll
    for (int e = 0; e < EPT; ++e) {
      p0 += th[e] * nT[e][0];  p1 += th[e] * nT[e][1];
      p2 += th[e] * woT[e][0]; p3 += th[e] * woT[e][1];
    }
    #pragma unroll
    for (int off = 16; off > 0; off >>= 1) {
      p0 += __shfl_xor(p0, off, 32); p1 += __shfl_xor(p1, off, 32);
      p2 += __shfl_xor(p2, off, 32); p3 += __shfl_xor(p3, off, 32);
    }
    if (lane == 0) {
      lds_red[0][wave][0] = p0; lds_red[0][wave][1] = p1;
      lds_red[0][wave][2] = p2; lds_red[0][wave][3] = p3;
    }
  }
  WAIT_ASYNC_LE(RING);   // input preload (8 oldest per thread) complete
  __syncthreads();       // publish lds_input + prologue partials

  // ---- 4) serial scan: one barrier per step, noise ring RING steps ahead ----
  for (int t = 0; t < SEQ; ++t) {
    const int p = t & 1;

    // finalize r for step t (from tanh of previous state); emit out[t-1]
    float r0 = 0.f, r1 = 0.f;
    #pragma unroll
    for (int w = 0; w < NWAVES; ++w) {
      r0 += lds_red[p][w][0];
      r1 += lds_red[p][w][1];
    }
    if (tid == 0 && t > 0) {
      float o0 = 0.f, o1 = 0.f;
      #pragma unroll
      for (int w = 0; w < NWAVES; ++w) {
        o0 += lds_red[p][w][2];
        o1 += lds_red[p][w][3];
      }
      outp[(t - 1) * OUT_SIZE + 0] = o0;
      outp[(t - 1) * OUT_SIZE + 1] = o1;
    }

    const float ix0 = sm.run.input[t * IN_SIZE + 0];
    const float ix1 = sm.run.input[t * IN_SIZE + 1];
    const float ix2 = sm.run.input[t * IN_SIZE + 2];
    const float ix3 = sm.run.input[t * IN_SIZE + 3];

    WAIT_ASYNC_LE(RING - 1);                          // oldest noise slab landed
    float4 nz = sm.run.noise[t & (RING - 1)][tid];    // this thread's own slot
    float nzv[EPT] = { nz.x, nz.y, nz.z, nz.w };

    #pragma unroll
    for (int e = 0; e < EPT; ++e) {
      float upd = r0 * mT[e][0] + r1 * mT[e][1]
                + ix0 * wiT[e][0] + ix1 * wiT[e][1]
                + ix2 * wiT[e][2] + ix3 * wiT[e][3];
      h[e] = (1.0f - ALPHA) * h[e] + NOISE_STD * nzv[e] + upd;
      th[e] = tanhf(h[e]);
    }

    // refill the slot we just consumed (RING steps ahead)
    if (t + RING < SEQ)
      async_copy_f4(noise + noise_base + (size_t)(t + RING) * HIDDEN,
                    (float*)&sm.run.noise[t & (RING - 1)][tid]);

    // fused partials: r for step t+1 and out for step t
    float p0 = 0.f, p1 = 0.f, p2 = 0.f, p3 = 0.f;
    #pragma unroll
    for (int e = 0; e < EPT; ++e) {
      p0 += th[e] * nT[e][0];  p1 += th[e] * nT[e][1];
      p2 += th[e] * woT[e][0]; p3 += th[e] * woT[e][1];
    }
    #pragma unroll
    for (int off = 16; off > 0; off >>= 1) {
      p0 += __shfl_xor(p0, off, 32); p1 += __shfl_xor(p1, off, 32);
      p2 += __shfl_xor(p2, off, 32); p3 += __shfl_xor(p3, off, 32);
    }
    if (lane == 0) {
      const int q = (t + 1) & 1;
      lds_red[q][wave][0] = p0; lds_red[q][wave][1] = p1;
      lds_red[q][wave][2] = p2; lds_red[q][wave][3] = p3;
    }
    __syncthreads();
  }

  // ---- 5) final output (t = SEQ-1) ----
  if (tid == 0) {
    const int p = SEQ & 1;
    float o0 = 0.f, o1 = 0.f;
    #pragma unroll
    for (int w = 0; w < NWAVES; ++w) {
      o0 += lds_red[p][w][2];
      o1 += lds_red[p][w][3];
    }
    outp[(SEQ - 1) * OUT_SIZE + 0] = o0;
    outp[(SEQ - 1) * OUT_SIZE + 1] = o1;
  }
}

extern "C" void kernel_launch(void* const* d_in, const int* in_sizes, int n_in,
                              void* d_out, int out_size, void* d_ws, size_t ws_size,
                              hipStream_t stream) {
  (void)in_sizes; (void)n_in; (void)out_size; (void)d_ws; (void)ws_size;
  const float* input    = (const float*)d_in[0];
  const float* noise    = (const float*)d_in[1];
  const float* basis    = (const float*)d_in[2];
  const float* supports = (const float*)d_in[3];
  const float* wi       = (const float*)d_in[4];
  const float* mp       = (const float*)d_in[5];
  const float* np       = (const float*)d_in[6];
  const float* wo       = (const float*)d_in[7];
  const float* h0       = (const float*)d_in[8];
  float* out = (float*)d_out;

  hipLaunchKernelGGL(SupportLowRankRNN_kernel, dim3(BATCH), dim3(THREADS), 0, stream,
                     input, noise, basis, supports, wi, mp, np, wo, h0, out);
}